// PolynomialFeatures_32023276159408
// MI455X (gfx1250) — compile-verified
//
#include <hip/hip_runtime.h>

// ---------------- problem constants (from reference) ----------------
#define BATCH   4096
#define NF      256
#define NPAIRS  32640                 // C(256,2), same_divider=1 keeps all pairs
#define OUTW    (NF + NPAIRS)         // 32896 floats per output row (16B-aligned stride)

// ---------------- tiling ----------------
#define ROWS    8                     // batch rows per block (x rows staged in LDS)
#define NCHUNK  4                     // pair-list split across gridDim.y
#define CHUNK   (NPAIRS / NCHUNK)     // 8160 pairs per chunk (multiple of 4)
#define GROUPS  (CHUNK / 4)           // 2040 float4 groups per chunk
#define TPB     256                   // 8 waves (wave32)

typedef float v4f __attribute__((ext_vector_type(4)));
typedef int   v4i __attribute__((ext_vector_type(4)));

__global__ __launch_bounds__(TPB) void poly_cross_kernel(
    const float* __restrict__ x,
    const int*   __restrict__ idx_a,
    const int*   __restrict__ idx_b,
    float*       __restrict__ out)
{
    __shared__ float xl[ROWS * NF];   // 8 KB: 8 staged batch rows

    const int tid = threadIdx.x;
    const int b0  = blockIdx.x * ROWS;
    const int c   = blockIdx.y;

    // ---- stage 8 x-rows (8 KB) into LDS via gfx1250 async global->LDS DMA ----
    // 512 float4 quads total; each thread issues exactly 2 async b128 loads.
    // Fully unrolled so no exec-mask loop machinery is generated.
    {
        const float* gbase = x + (size_t)b0 * NF;
        unsigned lds0 = (unsigned)(size_t)(&xl[tid * 4]);
        unsigned lds1 = (unsigned)(size_t)(&xl[(tid + TPB) * 4]);
        unsigned long long g0 = (unsigned long long)(size_t)(gbase + tid * 4);
        unsigned long long g1 = (unsigned long long)(size_t)(gbase + (tid + TPB) * 4);
        asm volatile("global_load_async_to_lds_b128 %0, %1, off"
                     :: "v"(lds0), "v"(g0) : "memory");
        asm volatile("global_load_async_to_lds_b128 %0, %1, off"
                     :: "v"(lds1), "v"(g1) : "memory");
        asm volatile("s_wait_asynccnt 0" ::: "memory");  // this wave's LDS writes landed
    }
    __syncthreads();                                     // all waves' writes visible

    // ---- pass-through copy of raw features (only chunk-0 blocks) ----
    if (c == 0) {
        #pragma unroll
        for (int r = 0; r < ROWS; ++r) {
            float v = xl[r * NF + tid];
            __builtin_nontemporal_store(v, out + (size_t)(b0 + r) * OUTW + tid);
        }
    }

    // ---- crossed features: gather from LDS, multiply, NT b128 stores ----
    const int p0 = c * CHUNK;
    for (int g = tid; g < GROUPS; g += TPB) {
        const int p = p0 + (g << 2);
        const v4i ia = *(const v4i*)(idx_a + p);      // contiguous, L2-resident
        const v4i ib = *(const v4i*)(idx_b + p);
        #pragma unroll
        for (int r = 0; r < ROWS; ++r) {
            const float* xr = &xl[r * NF];
            v4f v;
            v.x = xr[ia.x] * xr[ib.x];
            v.y = xr[ia.y] * xr[ib.y];
            v.z = xr[ia.z] * xr[ib.z];
            v.w = xr[ia.w] * xr[ib.w];
            // consecutive lanes -> consecutive float4s: coalesced 512B/wave,
            // non-temporal so the 539MB write stream doesn't thrash L2.
            __builtin_nontemporal_store(
                v, (v4f*)(out + (size_t)(b0 + r) * OUTW + NF + p));
        }
    }
}

extern "C" void kernel_launch(void* const* d_in, const int* in_sizes, int n_in,
                              void* d_out, int out_size, void* d_ws, size_t ws_size,
                              hipStream_t stream) {
    const float* x     = (const float*)d_in[0];
    const int*   idx_a = (const int*)d_in[1];
    const int*   idx_b = (const int*)d_in[2];
    float*       out   = (float*)d_out;

    dim3 grid(BATCH / ROWS, NCHUNK);   // (512, 4)
    dim3 block(TPB);
    poly_cross_kernel<<<grid, block, 0, stream>>>(x, idx_a, idx_b, out);
}